// HybridODENet_59150289600901
// MI455X (gfx1250) — compile-verified
//
#include <hip/hip_runtime.h>
#include <math.h>

#define N_BALLS 256
#define BATCH   128
#define MAX_EVENTS 5
#define HID     64
#define NPAIR   32640                    // 256*255/2
#define NCAND   (NPAIR + N_BALLS*4)
#define STATE_ELEMS (BATCH*N_BALLS*4)
#define WORLD_W 10.0f
#define WORLD_H 10.0f
#define INF_G   1e30f
#define NTHREADS 256

typedef __attribute__((ext_vector_type(2))) float v2f;
typedef __attribute__((ext_vector_type(8))) float v8f;

__device__ __forceinline__ float silu_f(float x) {
    return x / (1.0f + expf(-x));
}

// invert p -> (i,j), i<j, row-major triu order (np.triu_indices(256,1))
__device__ __forceinline__ void decode_pair(int p, int& io, int& jo) {
    float fp = (float)p;
    int ii = (int)floorf((511.0f - sqrtf(fmaxf(261121.0f - 8.0f*fp, 0.0f))) * 0.5f);
    if (ii < 0) ii = 0;
    if (ii > 254) ii = 254;
    while (ii > 0 && (ii*(511 - ii))/2 > p) --ii;
    while (ii < 254 && ((ii+1)*(511 - (ii+1)))/2 <= p) ++ii;
    io = ii;
    jo = ii + 1 + (p - (ii*(511 - ii))/2);
}

extern "C" __global__ __launch_bounds__(NTHREADS)
void hybrid_ode_rollout(const float* __restrict__ state0,
                        const float* __restrict__ radii,
                        const float* __restrict__ W1, const float* __restrict__ b1,
                        const float* __restrict__ W2, const float* __restrict__ b2,
                        const float* __restrict__ W3, const float* __restrict__ b3,
                        const float* __restrict__ dtp, const int* __restrict__ nstepsp,
                        float* __restrict__ out)
{
    const int t    = threadIdx.x;
    const int lane = t & 31;
    const int wave = t >> 5;

    __shared__ float s_w2[HID*HID];       // W2, row-major as input (B[k][n] = W2[n][k])
    __shared__ float s_h1[BATCH*HID];     // layer-1 activations [batch][hid]
    __shared__ float s_pos[N_BALLS*2];
    __shared__ float s_vel[N_BALLS*2];
    __shared__ float s_rad[N_BALLS];
    __shared__ float red_val[NTHREADS];
    __shared__ int   red_idx[NTHREADS];
    __shared__ float s_part[BATCH*4];     // per-(batch, n-tile) partial layer-3 sums
    __shared__ float s_nx[BATCH], s_ny[BATCH];
    __shared__ float s_w1[HID*2], s_b1[HID], s_b2[HID], s_w3[HID];
    __shared__ float sc_tc, sc_dte, sc_r, sc_wnx, sc_wny, sc_wp;
    __shared__ int   sc_isball, sc_i, sc_j, sc_iw, sc_casea, sc_caseb, sc_done;

    // stage constants
    for (int i=t; i<HID*HID; i+=NTHREADS) s_w2[i] = W2[i];
    for (int i=t; i<HID*2;   i+=NTHREADS) s_w1[i] = W1[i];
    for (int i=t; i<HID;     i+=NTHREADS) { s_b1[i]=b1[i]; s_b2[i]=b2[i]; s_w3[i]=W3[i]; }
    for (int i=t; i<N_BALLS; i+=NTHREADS) s_rad[i] = radii[i];
    // snapshot 0 = input state
    for (int i=t; i<STATE_ELEMS; i+=NTHREADS) out[i] = state0[i];

    const float dt      = dtp[0];
    const int   n_steps = nstepsp[0];
    const float fb3     = b3[0];
    __syncthreads();

    for (int step = 0; step < n_steps; ++step) {
        const float* prev = out + (size_t)step     * STATE_ELEMS;
        float*       cur  = out + (size_t)(step+1) * STATE_ELEMS;
        for (int i=t; i<STATE_ELEMS; i+=NTHREADS) cur[i] = prev[i];
        const float t_s = (float)step * dt;
        const float t_e = t_s + dt;
        if (t == 0) { sc_tc = t_s; sc_done = 0; }
        __syncthreads();

        for (int ev = 0; ev < MAX_EVENTS; ++ev) {
            if (sc_done) break;                     // uniform across block
            // batch-0 snapshot into LDS for detection
            for (int i=t; i<N_BALLS; i+=NTHREADS) {
                const float* s4 = cur + i*4;
                s_pos[2*i]=s4[0]; s_pos[2*i+1]=s4[1];
                s_vel[2*i]=s4[2]; s_vel[2*i+1]=s4[3];
            }
            __syncthreads();

            // candidate scan: 32640 pairs + 1024 walls
            float bestv = 3.0e38f; int besti = 0x7fffffff;
            for (int p=t; p<NCAND; p+=NTHREADS) {
                float g;
                if (p < NPAIR) {
                    int i, j; decode_pair(p, i, j);
                    float dx = s_pos[2*j]   - s_pos[2*i];
                    float dy = s_pos[2*j+1] - s_pos[2*i+1];
                    float dist = sqrtf(dx*dx + dy*dy);
                    float inv  = 1.0f / (dist + 1e-8f);
                    float app  = (s_vel[2*j]-s_vel[2*i])*dx*inv
                               + (s_vel[2*j+1]-s_vel[2*i+1])*dy*inv;
                    g = (app < 0.0f) ? (dist - (s_rad[i] + s_rad[j])) : INF_G;
                } else {
                    int kw = p - NPAIR; int bi = kw >> 2; int wid = kw & 3;
                    float r  = s_rad[bi];
                    float px = s_pos[2*bi], py = s_pos[2*bi+1];
                    float vx = s_vel[2*bi], vy = s_vel[2*bi+1];
                    float gw; bool mv;
                    if (wid == 0)      { gw = px - r;           mv = vx < 0.0f; }
                    else if (wid == 1) { gw = WORLD_W - px - r; mv = vx > 0.0f; }
                    else if (wid == 2) { gw = py - r;           mv = vy < 0.0f; }
                    else               { gw = WORLD_H - py - r; mv = vy > 0.0f; }
                    g = mv ? gw : INF_G;
                }
                if (g < bestv) { bestv = g; besti = p; }
            }
            red_val[t] = bestv; red_idx[t] = besti;
            __syncthreads();
            for (int s = NTHREADS/2; s > 0; s >>= 1) {
                if (t < s) {
                    float v2 = red_val[t+s]; int i2 = red_idx[t+s];
                    if (v2 < red_val[t] || (v2 == red_val[t] && i2 < red_idx[t])) {
                        red_val[t] = v2; red_idx[t] = i2;
                    }
                }
                __syncthreads();
            }

            if (t == 0) {
                float gap = red_val[0]; int k = red_idx[0];
                int is_ball = (k < NPAIR) ? 1 : 0;
                int i_b = 0, j_b = 1, i_w = 0, wid = 0;
                if (is_ball) decode_pair(k, i_b, j_b);
                else { int kw = k - NPAIR; i_w = kw >> 2; wid = kw & 3; }
                float app;
                if (is_ball) {
                    float nx = s_pos[2*j_b]   - s_pos[2*i_b];
                    float ny = s_pos[2*j_b+1] - s_pos[2*i_b+1];
                    float nn = sqrtf(nx*nx + ny*ny);
                    app = -(((s_vel[2*j_b]-s_vel[2*i_b])*nx
                           + (s_vel[2*j_b+1]-s_vel[2*i_b+1])*ny) / nn);
                } else {
                    app = fabsf(fmaxf(s_vel[2*i_w], s_vel[2*i_w+1]));
                }
                float t_c  = sc_tc;
                float t_ev = t_c + gap / fmaxf(app, 1e-6f);
                float dte  = (t_ev > t_c + 1e-10f) ? (t_ev - t_c) : 0.0f;
                int no_event = (gap > 0.05f) ? 1 : 0;
                int act    = (!sc_done) && (!no_event);
                int case_a = act && (gap <= 0.0f);
                int case_b = act && (gap > 0.0f) && (app > 1e-6f) && (t_ev < t_e);
                sc_done = sc_done || no_event ||
                          (act && (gap > 0.0f) && ((app <= 1e-6f) || (t_ev >= t_e)));
                if (case_b) sc_tc = t_ev;
                sc_isball = is_ball; sc_i = i_b; sc_j = j_b; sc_iw = i_w;
                sc_casea = case_a; sc_caseb = case_b; sc_dte = dte;
                sc_r = s_rad[i_w];
                const float wnx_t[4] = {1.f, -1.f, 0.f, 0.f};
                const float wny_t[4] = {0.f, 0.f, 1.f, -1.f};
                const float wp_t[4]  = {0.f, -WORLD_W, 0.f, -WORLD_H};
                sc_wnx = wnx_t[wid]; sc_wny = wny_t[wid]; sc_wp = wp_t[wid];
            }
            __syncthreads();

            const int casea = sc_casea, caseb = sc_caseb, isball = sc_isball;
            const float dte = sc_dte;
            if (casea || caseb) {
                if (caseb) {                      // state_b path: integrate all positions
                    for (int i=t; i<BATCH*N_BALLS; i+=NTHREADS) {
                        float* s4 = cur + i*4;
                        s4[0] += s4[2]*dte;
                        s4[1] += s4[3]*dte;
                    }
                }
                __syncthreads();
                if (isball) {
                    const int ib = sc_i, jb = sc_j;
                    if (t < BATCH) {              // per-batch features + nhat
                        const float* pi = cur + (t*N_BALLS + ib)*4;
                        const float* pj = cur + (t*N_BALLS + jb)*4;
                        float dx = pj[0]-pi[0], dy = pj[1]-pi[1];
                        float dist = fmaxf(sqrtf(dx*dx + dy*dy), 1e-8f);
                        float nx = dx/dist, ny = dy/dist;
                        float app = (pj[2]-pi[2])*nx + (pj[3]-pi[3])*ny;
                        red_val[t]       = dist;
                        red_val[BATCH+t] = app;
                        s_nx[t] = nx; s_ny[t] = ny;
                    }
                    __syncthreads();
                    // layer 1: h1 = silu(x @ W1^T + b1)   (128x2 @ 2x64)
                    for (int idx=t; idx<BATCH*HID; idx+=NTHREADS) {
                        int bb = idx >> 6, kk = idx & 63;
                        float x0 = red_val[bb], x1 = red_val[BATCH+bb];
                        s_h1[idx] = silu_f(x0*s_w1[2*kk] + x1*s_w1[2*kk+1] + s_b1[kk]);
                    }
                    __syncthreads();
                    // layer 2 via V_WMMA_F32_16X16X4_F32: h2 = silu(h1 @ W2^T + b2),
                    // fused with layer 3 (dot with W3) via deterministic lane reduction.
                    const int m15 = lane & 15;
                    const int khi = (lane >> 4) << 1;      // K offset 0 (lanes 0-15) / 2 (16-31)
                    for (int tile = wave; tile < 32; tile += 8) {
                        const int mt = tile & 7, nt = tile >> 3;
                        const int m  = mt*16 + m15;        // A row (batch)
                        const int nn = nt*16 + m15;        // B column (output unit)
                        v8f acc = {};
                        for (int kt = 0; kt < 16; ++kt) {
                            const int kb = kt*4 + khi;
                            v2f av, bv;
                            av.x = s_h1[m*HID + kb];       // A[m][K], K=khi
                            av.y = s_h1[m*HID + kb + 1];   // A[m][K+1]
                            bv.x = s_w2[nn*HID + kb];      // B[K][n] = W2[n][K]
                            bv.y = s_w2[nn*HID + kb + 1];
                            acc = __builtin_amdgcn_wmma_f32_16x16x4_f32(
                                      false, av, false, bv, (short)0, acc, false, false);
                        }
                        const float bias = s_b2[nn];
                        const float w3n  = s_w3[nn];
                        const int mbase  = mt*16 + ((lane >> 4) << 3);
                        #pragma unroll
                        for (int v = 0; v < 8; ++v) {
                            float c = silu_f(acc[v] + bias) * w3n;   // layer-3 contribution
                            c += __shfl_xor(c, 1);
                            c += __shfl_xor(c, 2);
                            c += __shfl_xor(c, 4);
                            c += __shfl_xor(c, 8);        // sum over 16 n-columns
                            if ((lane & 15) == 0)
                                s_part[(mbase + v)*4 + nt] = c;
                        }
                    }
                    __syncthreads();
                    if (t < BATCH) {                      // impulse -> velocity update
                        float imp = s_part[t*4+0] + s_part[t*4+1]
                                  + s_part[t*4+2] + s_part[t*4+3] + fb3;
                        float ix = imp * s_nx[t], iy = imp * s_ny[t];
                        float* vi = cur + (t*N_BALLS + ib)*4 + 2;
                        float* vj = cur + (t*N_BALLS + jb)*4 + 2;
                        vi[0] += ix; vi[1] += iy;
                        vj[0] -= ix; vj[1] -= iy;
                    }
                } else {                                  // wall reflection
                    if (t < BATCH) {
                        const int iw = sc_iw;
                        const float wnx = sc_wnx, wny = sc_wny, wp = sc_wp, r = sc_r;
                        float* s4 = cur + (t*N_BALLS + iw)*4;
                        float px=s4[0], py=s4[1], vx=s4[2], vy=s4[3];
                        float vn = vx*wnx + vy*wny;
                        vx -= 2.0f*vn*wnx; vy -= 2.0f*vn*wny;
                        float pn  = px*wnx + py*wny;
                        float pen = fmaxf(wp + r - pn, 0.0f);
                        px += pen*wnx; py += pen*wny;
                        s4[0]=px; s4[1]=py; s4[2]=vx; s4[3]=vy;
                    }
                }
            }
            __syncthreads();
        } // events

        // integrate to end of step
        const float tcf  = sc_tc;
        const float dtef = (t_e > tcf + 1e-10f) ? (t_e - tcf) : 0.0f;
        for (int i=t; i<BATCH*N_BALLS; i+=NTHREADS) {
            float* s4 = cur + i*4;
            s4[0] += s4[2]*dtef;
            s4[1] += s4[3]*dtef;
        }
        __syncthreads();
    }
}

extern "C" void kernel_launch(void* const* d_in, const int* in_sizes, int n_in,
                              void* d_out, int out_size, void* d_ws, size_t ws_size,
                              hipStream_t stream) {
    (void)in_sizes; (void)n_in; (void)out_size; (void)d_ws; (void)ws_size;
    const float* state = (const float*)d_in[0];
    const float* radii = (const float*)d_in[1];
    const float* W1    = (const float*)d_in[2];
    const float* b1    = (const float*)d_in[3];
    const float* W2    = (const float*)d_in[4];
    const float* b2    = (const float*)d_in[5];
    const float* W3    = (const float*)d_in[6];
    const float* b3    = (const float*)d_in[7];
    const float* dtp   = (const float*)d_in[8];
    const int*   nst   = (const int*)d_in[9];
    float* out = (float*)d_out;
    hipLaunchKernelGGL(hybrid_ode_rollout, dim3(1), dim3(NTHREADS), 0, stream,
                       state, radii, W1, b1, W2, b2, W3, b3, dtp, nst, out);
}